// GIN_63333587746870
// MI455X (gfx1250) — compile-verified
//
#include <hip/hip_runtime.h>
#include <hip/hip_bf16.h>

typedef float v2f __attribute__((ext_vector_type(2)));
typedef float v8f __attribute__((ext_vector_type(8)));

#define HCH 64          // channels everywhere (IN=H=OUT=64)
#define GG  512         // graphs
#define STAT_ROWS 512   // rows per column-stat partial block

// ---------------------------------------------------------------------------
// WMMA fp32 GEMM:  Y[M,64] = A[M,64] @ W[64,64] + bias   (optionally += old Y)
// One wave32 per 16-row tile; 4 column tiles of 16; K=64 as 16 x (K=4) WMMA.
// A-frag:  lane m=lane&15 holds rows, k-pair selected by lane>>4.
// B-frag:  lane n=lane&15 holds cols, same k-pair split (mirror of A layout).
// C/D:     vgpr r -> row r + 8*(lane>>4), col = lane&15.
//
// W is staged into LDS in *fragment-packed* layout so each B-fragment is one
// contiguous 8-byte ds_load_b64:
//   sWp[(p*64 + n)*2 + j] = W[2p + j][n],  p = k-pair index 0..31, j = 0..1
// ---------------------------------------------------------------------------
__global__ void __launch_bounds__(256)
k_gemm64(const float* __restrict__ A, const float* __restrict__ W,
         const float* __restrict__ bias, float* __restrict__ Y,
         int M, int accumulate) {
    __shared__ alignas(16) float sWp[HCH * HCH];   // packed fragments, 16 KB
    __shared__ float sB[HCH];
    for (int i = threadIdx.x; i < HCH * HCH; i += 256) {
        const int k = i >> 6;                      // W row
        const int n = i & 63;                      // W col
        sWp[(((k >> 1) * HCH) + n) * 2 + (k & 1)] = W[i];
    }
    if (threadIdx.x < HCH) sB[threadIdx.x] = bias[threadIdx.x];
    __syncthreads();

    const int wave = threadIdx.x >> 5;
    const int lane = threadIdx.x & 31;
    const int tile = blockIdx.x * 8 + wave;
    if (tile * 16 >= M) return;                    // whole-wave uniform exit

    const int row0 = tile * 16;
    const int m  = lane & 15;                      // A-row / B-col / D-col in tile
    const int kh = lane >> 4;                      // k-pair half select
    const float* __restrict__ Arow = A + (size_t)(row0 + m) * HCH;

    v8f acc[4];
#pragma unroll
    for (int nt = 0; nt < 4; ++nt) {
        const float bv = sB[nt * 16 + m];          // bias broadcast: col-only
#pragma unroll
        for (int r = 0; r < 8; ++r) acc[nt][r] = bv;
    }

#pragma unroll
    for (int ks = 0; ks < 16; ++ks) {
        const int k0 = ks * 4 + kh * 2;            // first K of this lane's pair
        const int p  = ks * 2 + kh;                // k-pair index in packed LDS
        v2f a;
        a.x = Arow[k0];
        a.y = Arow[k0 + 1];
#pragma unroll
        for (int nt = 0; nt < 4; ++nt) {
            const int n = nt * 16 + m;
            const v2f b = *(const v2f*)&sWp[(p * HCH + n) * 2];  // one ds_load_b64
            acc[nt] = __builtin_amdgcn_wmma_f32_16x16x4_f32(
                false, a, false, b, (short)0, acc[nt], false, false);
        }
    }

#pragma unroll
    for (int nt = 0; nt < 4; ++nt) {
#pragma unroll
        for (int r = 0; r < 8; ++r) {
            const size_t idx = (size_t)(row0 + r + 8 * kh) * HCH + nt * 16 + m;
            if (accumulate) Y[idx] += acc[nt][r];
            else            Y[idx]  = acc[nt][r];
        }
    }
}

// ---------------------------------------------------------------------------
// Deterministic BatchNorm stats (no float atomics):
// pass 1: each block reduces STAT_ROWS rows -> per-column partial sum / sumsq
// pass 2: fixed-order reduce over blocks -> stats[c]=mu, stats[64+c]=rsqrt(var+eps)
// ---------------------------------------------------------------------------
__global__ void k_colstat_partial(const float* __restrict__ Y, float* __restrict__ psum,
                                  float* __restrict__ psq, int M) {
    const int c  = threadIdx.x;                    // 64 threads, coalesced over c
    const int r0 = blockIdx.x * STAT_ROWS;
    int r1 = r0 + STAT_ROWS; if (r1 > M) r1 = M;
    float s = 0.f, q = 0.f;
    for (int r = r0; r < r1; ++r) {
        const float v = Y[(size_t)r * HCH + c];
        s += v; q += v * v;
    }
    psum[blockIdx.x * HCH + c] = s;
    psq [blockIdx.x * HCH + c] = q;
}

__global__ void k_colstat_final(const float* __restrict__ psum, const float* __restrict__ psq,
                                int nblk, float invM, float* __restrict__ stats) {
    const int c = threadIdx.x;                     // 64 threads
    float s = 0.f, q = 0.f;
    for (int b = 0; b < nblk; ++b) { s += psum[b * HCH + c]; q += psq[b * HCH + c]; }
    const float mu  = s * invM;
    const float var = q * invM - mu * mu;
    stats[c]        = mu;
    stats[HCH + c]  = rsqrtf(var + 1e-5f);
}

__global__ void k_bnrelu(const float* __restrict__ Y, const float* __restrict__ stats,
                         const float* __restrict__ g, const float* __restrict__ bt,
                         float* __restrict__ H, int total) {
    const int i = blockIdx.x * 256 + threadIdx.x;
    if (i >= total) return;
    const int c = i & (HCH - 1);
    const float v = (Y[i] - stats[c]) * stats[HCH + c] * g[c] + bt[c];
    H[i] = v > 0.f ? v : 0.f;
}

// z = (1 + ceps[l]) * h   (scatter target init)
__global__ void k_initz(const float* __restrict__ Hin, const float* __restrict__ ceps, int l,
                        float* __restrict__ Z, int total) {
    const int i = blockIdx.x * 256 + threadIdx.x;
    if (i >= total) return;
    Z[i] = (1.0f + ceps[l]) * Hin[i];
}

// z[dst] += h[src], 16 threads/edge, float4 gather + 4 f32 global atomics
__global__ void k_scatter(const float* __restrict__ H, const int* __restrict__ src,
                          const int* __restrict__ dst, float* __restrict__ Z, int E_) {
    const long long t = (long long)blockIdx.x * 256 + threadIdx.x;
    if (t >= (long long)E_ * 16) return;
    const int e = (int)(t >> 4);
    const int q = ((int)t & 15) * 4;
    const int s = src[e], d = dst[e];
    const float4 v = *(const float4*)(H + (size_t)s * HCH + q);
    float* p = Z + (size_t)d * HCH + q;
    atomicAdd(p + 0, v.x); atomicAdd(p + 1, v.y);
    atomicAdd(p + 2, v.z); atomicAdd(p + 3, v.w);
}

// Deterministic mean-pool: batch is sorted, so graph g owns a contiguous row
// range found by binary search. One block per graph, one thread per channel.
__device__ __forceinline__ int lb(const int* __restrict__ a, int n, int key) {
    int lo = 0, hi = n;
    while (lo < hi) { const int mid = (lo + hi) >> 1; if (a[mid] < key) lo = mid + 1; else hi = mid; }
    return lo;
}
__global__ void k_pool(const float* __restrict__ H, const int* __restrict__ batch,
                       float* __restrict__ P, int Nn) {
    const int g = blockIdx.x;                      // 512 blocks
    const int c = threadIdx.x;                     // 64 threads
    const int lo = lb(batch, Nn, g);
    const int hi = lb(batch, Nn, g + 1);
    float s = 0.f;
    for (int r = lo; r < hi; ++r) s += H[(size_t)r * HCH + c];
    const int cnt = hi - lo;
    P[g * HCH + c] = s / (float)(cnt > 0 ? cnt : 1);
}

// ---------------------------------------------------------------------------
static void mlp_half(const float* A, const float* W, const float* b,
                     const float* g, const float* bt,
                     float* y, float* hout, float* psum, float* psq, float* stats,
                     int Nn, hipStream_t stream) {
    const int mtiles = Nn / 16;
    k_gemm64<<<(mtiles + 7) / 8, 256, 0, stream>>>(A, W, b, y, Nn, 0);
    const int nblk = (Nn + STAT_ROWS - 1) / STAT_ROWS;
    k_colstat_partial<<<nblk, HCH, 0, stream>>>(y, psum, psq, Nn);
    k_colstat_final<<<1, HCH, 0, stream>>>(psum, psq, nblk, 1.0f / (float)Nn, stats);
    k_bnrelu<<<(Nn * HCH + 255) / 256, 256, 0, stream>>>(y, stats, g, bt, hout, Nn * HCH);
}

extern "C" void kernel_launch(void* const* d_in, const int* in_sizes, int n_in,
                              void* d_out, int out_size, void* d_ws, size_t ws_size,
                              hipStream_t stream) {
    const float* x     = (const float*)d_in[0];
    const int*   ei    = (const int*)  d_in[1];
    const int*   batch = (const int*)  d_in[2];
    const float* fW1   = (const float*)d_in[3];
    const float* fb1   = (const float*)d_in[4];
    const float* fg1   = (const float*)d_in[5];
    const float* fbt1  = (const float*)d_in[6];
    const float* fW2   = (const float*)d_in[7];
    const float* fb2   = (const float*)d_in[8];
    const float* fg2   = (const float*)d_in[9];
    const float* fbt2  = (const float*)d_in[10];
    const float* cW1   = (const float*)d_in[11];
    const float* cb1   = (const float*)d_in[12];
    const float* cg1   = (const float*)d_in[13];
    const float* cbt1  = (const float*)d_in[14];
    const float* cW2   = (const float*)d_in[15];
    const float* cb2   = (const float*)d_in[16];
    const float* cg2   = (const float*)d_in[17];
    const float* cbt2  = (const float*)d_in[18];
    const float* ceps  = (const float*)d_in[19];
    const float* linW  = (const float*)d_in[20];
    const float* linb  = (const float*)d_in[21];

    const int Nn = in_sizes[0] / HCH;              // 50000
    const int E_ = in_sizes[1] / 2;                // 800000
    float* out = (float*)d_out;                    // [512, 64]

    // workspace layout (floats)
    float* y      = (float*)d_ws;                  // N*64
    float* h      = y + (size_t)Nn * HCH;          // N*64
    float* z      = h + (size_t)Nn * HCH;          // N*64
    float* pooled = z + (size_t)Nn * HCH;          // 512*64
    float* psum   = pooled + GG * HCH;             // nblk*64
    const int nblk = (Nn + STAT_ROWS - 1) / STAT_ROWS;
    float* psq    = psum + (size_t)nblk * HCH;
    float* stats  = psq + (size_t)nblk * HCH;      // 128

    // layer 0: first_h MLP
    mlp_half(x, fW1, fb1, fg1, fbt1, y, h, psum, psq, stats, Nn, stream);
    mlp_half(h, fW2, fb2, fg2, fbt2, y, h, psum, psq, stats, Nn, stream);
    // head 0 (fresh write -> idempotent across graph replays)
    k_pool<<<GG, HCH, 0, stream>>>(h, batch, pooled, Nn);
    k_gemm64<<<(GG / 16 + 7) / 8, 256, 0, stream>>>(pooled, linW, linb, out, GG, 0);

    // layers 1..3: GINConv
    for (int l = 0; l < 3; ++l) {
        k_initz<<<(Nn * HCH + 255) / 256, 256, 0, stream>>>(h, ceps, l, z, Nn * HCH);
        const long long sthreads = (long long)E_ * 16;
        k_scatter<<<(unsigned)((sthreads + 255) / 256), 256, 0, stream>>>(h, ei, ei + E_, z, E_);
        mlp_half(z, cW1 + (size_t)l * HCH * HCH, cb1 + l * HCH, cg1 + l * HCH, cbt1 + l * HCH,
                 y, h, psum, psq, stats, Nn, stream);
        mlp_half(h, cW2 + (size_t)l * HCH * HCH, cb2 + l * HCH, cg2 + l * HCH, cbt2 + l * HCH,
                 y, h, psum, psq, stats, Nn, stream);
        k_pool<<<GG, HCH, 0, stream>>>(h, batch, pooled, Nn);
        k_gemm64<<<(GG / 16 + 7) / 8, 256, 0, stream>>>(pooled,
                 linW + (size_t)(l + 1) * HCH * HCH, linb + (l + 1) * HCH, out, GG, 1);
    }
}